// DGNN_4801773437364
// MI455X (gfx1250) — compile-verified
//
#include <hip/hip_runtime.h>
#include <hip/hip_bf16.h>

typedef float v2f __attribute__((ext_vector_type(2)));
typedef float v8f __attribute__((ext_vector_type(8)));

#define NEG_SLOPE 0.01f
#define BN_EPS 1e-5f

// ---------------------------------------------------------------- zero
__global__ void zero_f4_kernel(float4* __restrict__ p, int n4) {
    int i = blockIdx.x * blockDim.x + threadIdx.x;
    if (i < n4) p[i] = make_float4(0.f, 0.f, 0.f, 0.f);
}

// ---------------------------------------------------------------- prep:
// zero-padded weight/bias copies + folded BN scale/shift (tiny, once per call)
__global__ void prep_kernel(const float* __restrict__ fc1W, const float* __restrict__ fc1b,
                            const float* __restrict__ fc2W, const float* __restrict__ fc2b,
                            const float* __restrict__ g, const float* __restrict__ b,
                            const float* __restrict__ m, const float* __restrict__ v,
                            float* __restrict__ fc1Wp, float* __restrict__ fc1bp,
                            float* __restrict__ fc2Wp, float* __restrict__ fc2bp,
                            float* __restrict__ bnsc, float* __restrict__ bnsh) {
    int i = blockIdx.x * blockDim.x + threadIdx.x;
    if (i < 128 * 96) {                       // fc1_W [128,90] -> [128,96]
        int r = i / 96, c = i - r * 96;
        fc1Wp[i] = (c < 90) ? fc1W[r * 90 + c] : 0.f;
        return;
    }
    int j = i - 128 * 96;
    if (j < 96 * 48) {                        // fc2_W [90,40] -> [96,48]
        int r = j / 48, c = j - r * 48;
        fc2Wp[j] = (r < 90 && c < 40) ? fc2W[r * 40 + c] : 0.f;
        return;
    }
    j -= 96 * 48;
    if (j < 96)  { fc1bp[j] = (j < 90) ? fc1b[j] : 0.f; return; }
    j -= 96;
    if (j < 48)  { fc2bp[j] = (j < 40) ? fc2b[j] : 0.f; return; }
    j -= 48;
    if (j < 128) {                            // BN fold: y = x*sc + sh
        float sc = g[j] * rsqrtf(v[j] + BN_EPS);
        bnsc[j] = sc;
        bnsh[j] = b[j] - m[j] * sc;
    }
}

// ---------------------------------------------------------------- edge coeff
__global__ void coeff_kernel(const int* __restrict__ dstI,
                             const float* __restrict__ edge_time,
                             const float* __restrict__ node_time,
                             const float* __restrict__ edge_weight,
                             float* __restrict__ coeff, int E) {
    int e = blockIdx.x * blockDim.x + threadIdx.x;
    if (e >= E) return;
    float dt = node_time[dstI[e]] - edge_time[e];
    float r  = dt > 0.f ? dt : 0.f;
    coeff[e] = edge_weight[e] * __expf(-r);
}

// ---------------------------------------------------------------- scatter
// one thread per (edge, 4 channels): float4 gather, 4 atomic f32 adds
template <int LOGC>
__global__ void scatter_kernel(const float* __restrict__ feat,
                               const float* __restrict__ coeff,
                               const int* __restrict__ srcI,
                               const int* __restrict__ dstI,
                               float* __restrict__ agg, int E) {
    const int C  = 1 << LOGC;
    const int TQ = C / 4;                       // threads per edge
    int idx = blockIdx.x * blockDim.x + threadIdx.x;   // < E*C/4 <= 51.2M
    int e = idx >> (LOGC - 2);
    if (e >= E) return;
    int q  = idx & (TQ - 1);
    int s  = srcI[e];
    int d  = dstI[e];
    float k = coeff[e];
    float4 f = ((const float4*)(feat + s * C))[q];
    float* ap = agg + d * C + q * 4;
    atomicAdd(ap + 0, f.x * k);
    atomicAdd(ap + 1, f.y * k);
    atomicAdd(ap + 2, f.z * k);
    atomicAdd(ap + 3, f.w * k);
}

// ---------------------------------------------------------------- WMMA f32 GEMM
// Cout[N, NCSTORE] = post( A[N,K] @ Wp[K,NCPAD] + biasP )
// MT row tiles (16 rows each) per block; one B fragment feeds MT WMMAs.
// POST: 0 = bias, 1 = bias+leaky, 2 = bias+BN+leaky
template <int K, int NCPAD, int MT, int POST, int NCSTORE>
__global__ __launch_bounds__(32 * (NCPAD / 16))
void gemm_wmma_kernel(const float* __restrict__ A,
                      const float* __restrict__ Wp,
                      const float* __restrict__ biasP,
                      const float* __restrict__ bnsc,
                      const float* __restrict__ bnsh,
                      float* __restrict__ Cout, int N) {
    constexpr int BD  = 32 * (NCPAD / 16);
    constexpr int MR  = 16 * MT;
    constexpr int LDA = K + 4;                  // 16B-aligned rows, conflict-free frags
    constexpr int KQ  = K / 4;
    __shared__ float Alds[MR * LDA];

    const int rowBase = blockIdx.x * MR;
    const int tid = threadIdx.x;
    const bool fullTile = (rowBase + MR <= N);  // true for all but the last block

    // cooperative float4 A-tile staging
    if (fullTile) {
#pragma unroll
        for (int i = tid; i < MR * KQ; i += BD) {
            int r = i / KQ, q = i - r * KQ;
            *(float4*)&Alds[r * LDA + q * 4] =
                ((const float4*)(A + (rowBase + r) * K))[q];
        }
    } else {
        for (int i = tid; i < MR * KQ; i += BD) {
            int r = i / KQ, q = i - r * KQ;
            int row = rowBase + r;
            float4 v = make_float4(0.f, 0.f, 0.f, 0.f);
            if (row < N) v = ((const float4*)(A + row * K))[q];
            *(float4*)&Alds[r * LDA + q * 4] = v;
        }
    }
    __syncthreads();

    const int lane = tid & 31;
    const int half = lane >> 4;                 // 0: lanes 0-15, 1: lanes 16-31
    const int lm   = lane & 15;
    const int col  = (tid >> 5) * 16 + lm;

    v8f acc[MT];
#pragma unroll
    for (int t = 0; t < MT; ++t) acc[t] = (v8f){0.f,0.f,0.f,0.f,0.f,0.f,0.f,0.f};

    const float* wb = Wp + col;
    for (int k0 = 0; k0 < K; k0 += 4) {
        int ka = k0 + 2 * half;                 // frag v0 K-index for this half
        v2f b;
        b.x = wb[ ka      * NCPAD];             // unconditional: Wp pre-padded
        b.y = wb[(ka + 1) * NCPAD];
#pragma unroll
        for (int t = 0; t < MT; ++t) {
            v2f a;
            a.x = Alds[(t * 16 + lm) * LDA + ka];
            a.y = Alds[(t * 16 + lm) * LDA + ka + 1];
            acc[t] = __builtin_amdgcn_wmma_f32_16x16x4_f32(
                false, a, false, b, (short)0, acc[t], false, false);
        }
    }

    float bcol = biasP[col];
    float sc = 1.f, sh = 0.f;
    if (POST == 2) { sc = bnsc[col]; sh = bnsh[col]; }

    // single hoisted column predicate (compile-time no-op unless NCSTORE<NCPAD)
    const bool colOk = (NCSTORE == NCPAD) || (col < NCSTORE);
    if (colOk) {
        float* cp = Cout + col;
        if (fullTile) {
#pragma unroll
            for (int t = 0; t < MT; ++t) {
#pragma unroll
                for (int r = 0; r < 8; ++r) {   // D: VGPR r -> row r + 8*half
                    int row = rowBase + t * 16 + half * 8 + r;
                    float v = acc[t][r] + bcol;
                    if (POST == 2) { v = v * sc + sh; v = (v >= 0.f) ? v : NEG_SLOPE * v; }
                    else if (POST == 1) { v = (v >= 0.f) ? v : NEG_SLOPE * v; }
                    cp[row * NCSTORE] = v;
                }
            }
        } else {
            for (int t = 0; t < MT; ++t) {
                for (int r = 0; r < 8; ++r) {
                    int row = rowBase + t * 16 + half * 8 + r;
                    float v = acc[t][r] + bcol;
                    if (POST == 2) { v = v * sc + sh; v = (v >= 0.f) ? v : NEG_SLOPE * v; }
                    else if (POST == 1) { v = (v >= 0.f) ? v : NEG_SLOPE * v; }
                    if (row < N) cp[row * NCSTORE] = v;
                }
            }
        }
    }
}

// ---------------------------------------------------------------- launcher
extern "C" void kernel_launch(void* const* d_in, const int* in_sizes, int n_in,
                              void* d_out, int out_size, void* d_ws, size_t ws_size,
                              hipStream_t stream) {
    const float* x           = (const float*)d_in[0];
    const int*   edge_index  = (const int*)  d_in[1];
    const float* edge_time   = (const float*)d_in[2];
    const float* node_time   = (const float*)d_in[3];
    const float* edge_weight = (const float*)d_in[4];
    const float* W1   = (const float*)d_in[5];
    const float* b1   = (const float*)d_in[6];
    const float* W2   = (const float*)d_in[7];
    const float* b2   = (const float*)d_in[8];
    const float* bn_g = (const float*)d_in[9];
    const float* bn_b = (const float*)d_in[10];
    const float* bn_m = (const float*)d_in[11];
    const float* bn_v = (const float*)d_in[12];
    const float* fc1W = (const float*)d_in[13];
    const float* fc1b = (const float*)d_in[14];
    const float* fc2W = (const float*)d_in[15];
    const float* fc2b = (const float*)d_in[16];

    const int N = in_sizes[0] / 64;    // 100000
    const int E = in_sizes[1] / 2;     // 1600000
    const int* srcI = edge_index;
    const int* dstI = edge_index + E;

    char*  ws  = (char*)d_ws;
    size_t off = 0;
    auto alloc = [&](size_t bytes) -> float* {
        float* p = (float*)(ws + off);
        off += (bytes + 255) & ~size_t(255);
        return p;
    };
    float* agg1  = alloc((size_t)N * 64  * 4);  // adjacent to agg2 (single zero pass)
    float* agg2  = alloc((size_t)N * 128 * 4);
    float* coeff = alloc((size_t)E * 4);
    float* h1    = alloc((size_t)N * 128 * 4);
    float* h3    = alloc((size_t)N * 128 * 4);
    float* h4    = alloc((size_t)N * 96  * 4);
    float* fc1Wp = alloc(128 * 96 * 4);
    float* fc2Wp = alloc(96 * 48 * 4);
    float* fc1bp = alloc(96 * 4);
    float* fc2bp = alloc(48 * 4);
    float* bnsc  = alloc(128 * 4);
    float* bnsh  = alloc(128 * 4);

    // 1) zero accumulators (agg1|agg2 contiguous: 192 floats/node)
    {
        int n4 = N * 192 / 4;
        zero_f4_kernel<<<(n4 + 255) / 256, 256, 0, stream>>>((float4*)agg1, n4);
    }
    // 2) pad weights/bias, fold BN
    {
        int total = 128 * 96 + 96 * 48 + 96 + 48 + 128;
        prep_kernel<<<(total + 255) / 256, 256, 0, stream>>>(
            fc1W, fc1b, fc2W, fc2b, bn_g, bn_b, bn_m, bn_v,
            fc1Wp, fc1bp, fc2Wp, fc2bp, bnsc, bnsh);
    }
    // 3) per-edge time-decay coefficient
    coeff_kernel<<<(E + 255) / 256, 256, 0, stream>>>(dstI, edge_time, node_time,
                                                      edge_weight, coeff, E);
    // 4) layer-1 scatter (64 ch, 16 threads/edge)
    {
        long long t = (long long)E * 16;
        scatter_kernel<6><<<(unsigned)((t + 255) / 256), 256, 0, stream>>>(
            x, coeff, srcI, dstI, agg1, E);
    }
    const int g64 = (N + 63) / 64;              // 1563 blocks of 64 rows
    // 5) h1 = agg1 @ W1 + b1
    gemm_wmma_kernel<64, 128, 4, 0, 128><<<g64, 256, 0, stream>>>(
        agg1, W1, b1, nullptr, nullptr, h1, N);
    // 6) layer-2 scatter (128 ch, 32 threads/edge -> wave-uniform e)
    {
        long long t = (long long)E * 32;
        scatter_kernel<7><<<(unsigned)((t + 255) / 256), 256, 0, stream>>>(
            h1, coeff, srcI, dstI, agg2, E);
    }
    // 7) h3 = leaky(BN(agg2 @ W2 + b2))
    gemm_wmma_kernel<128, 128, 4, 2, 128><<<g64, 256, 0, stream>>>(
        agg2, W2, b2, bnsc, bnsh, h3, N);
    // 8) h4 = leaky(h3 @ fc1Wp + fc1bp)   [N,96] (cols 90..95 are zeros)
    gemm_wmma_kernel<128, 96, 4, 1, 96><<<g64, 192, 0, stream>>>(
        h3, fc1Wp, fc1bp, nullptr, nullptr, h4, N);
    // 9) out = h4 @ fc2Wp + fc2bp         store first 40 of 48 cols
    gemm_wmma_kernel<96, 48, 4, 0, 40><<<g64, 96, 0, stream>>>(
        h4, fc2Wp, fc2bp, nullptr, nullptr, (float*)d_out, N);
}